// CDR_75642964017548
// MI455X (gfx1250) — compile-verified
//
#include <hip/hip_runtime.h>
#include <math.h>

typedef float v2f __attribute__((ext_vector_type(2)));
typedef float v4f __attribute__((ext_vector_type(4)));
typedef float v8f __attribute__((ext_vector_type(8)));

#define ALPHA 0.005f
// P=(1,2), T=(1,2): xd = L1/1 + L2/2

// ---- tiling constants ----
constexpr int Dd  = 1024;  // feature dim
constexpr int Od  = 256;   // centroids
constexpr int BT  = 16;    // rows per block
constexpr int WT  = 16;    // cols per wave tile
constexpr int NW  = 8;     // waves per block (256 threads, wave32)
constexpr int CT  = NW * WT;   // 128 cols per block
constexpr int KC  = 16;    // K chunk per LDS stage
constexpr int PAD = 20;    // padded LDS row (floats) -> conflict-free float4 reads

// Fused kernel: xd[b,o] = L1(b,o) + 0.5*sqrt(max(||x||^2+||c||^2-2*dot, 0))
// dot computed on the matrix pipe via V_WMMA_F32_16X16X4_F32.
__global__ __launch_bounds__(256) void CDR_dist_wmma_kernel(
    const float* __restrict__ x, const float* __restrict__ basis,
    float* __restrict__ xd)
{
    __shared__ float sNB[CT];            // ||c||^2 for block's 128 cols
    __shared__ float sNX[BT];            // ||x||^2 for block's 16 rows
    __shared__ float sX[BT][PAD];        // x K-tile (shared by all waves)
    __shared__ float sB[NW][WT][PAD];    // per-wave basis K-tile

    const int t    = threadIdx.x;
    const int wave = t >> 5;
    const int lane = t & 31;
    const int hi   = lane >> 4;          // 0 | 1 (half-wave)
    const int lo   = lane & 15;
    const int b0   = blockIdx.y * BT;
    const int o0   = blockIdx.x * CT;
    const int ow   = o0 + wave * WT;     // this wave's column base

    // ---- prologue: norms (tiny vs. main loop) ----
    if (t < CT) {
        const float* p = basis + (size_t)(o0 + t) * Dd;
        float s = 0.f;
        #pragma unroll 4
        for (int d = 0; d < Dd; d += 4) {
            v4f v = *(const v4f*)(p + d);
            s += v.x*v.x + v.y*v.y + v.z*v.z + v.w*v.w;
        }
        sNB[t] = s;
    } else if (t < CT + BT) {
        const float* p = x + (size_t)(b0 + (t - CT)) * Dd;
        float s = 0.f;
        #pragma unroll 4
        for (int d = 0; d < Dd; d += 4) {
            v4f v = *(const v4f*)(p + d);
            s += v.x*v.x + v.y*v.y + v.z*v.z + v.w*v.w;
        }
        sNX[t - CT] = s;
    }

    v8f c = {0.f,0.f,0.f,0.f,0.f,0.f,0.f,0.f};   // WMMA accumulator (dot)
    // 4 independent scalar partial sums per owned row: the accumulate is forced
    // into v_add_f32_e64 with a free |src| modifier (VOP3P pk_add has no abs).
    float l1a[8], l1b[8], l1c[8], l1d[8];
    #pragma unroll
    for (int i = 0; i < 8; i++) { l1a[i] = 0.f; l1b[i] = 0.f; l1c[i] = 0.f; l1d[i] = 0.f; }

    for (int k0 = 0; k0 < Dd; k0 += KC) {
        __syncthreads();   // previous chunk fully consumed before overwrite

        // stage x tile [16][16]: 256 floats, one per thread, coalesced per row
        {
            const int r = t >> 4, kk = t & 15;
            sX[r][kk] = x[(size_t)(b0 + r) * Dd + (k0 + kk)];
        }
        // stage per-wave basis tile [16][16]: 8 floats per lane (2x float4)
        {
            const int r  = lane >> 1;
            const int kk = (lane & 1) * 8;
            const float* p = basis + (size_t)(ow + r) * Dd + (k0 + kk);
            *(v4f*)&sB[wave][r][kk]     = *(const v4f*)(p);
            *(v4f*)&sB[wave][r][kk + 4] = *(const v4f*)(p + 4);
        }
        __syncthreads();

        // ---- matrix pipe: dot-product via f32 WMMA, K=4 per instruction ----
        // A (16x4): lanes 0-15 -> M=lane, K={0,1}; lanes 16-31 -> M=lane-16, K={2,3}
        // B (4x16): lanes 0-15 -> N=lane, K={0,1}; lanes 16-31 -> N=lane-16, K={2,3}
        #pragma unroll
        for (int kk = 0; kk < KC; kk += 4) {
            const int ks = kk + hi * 2;
            v2f a, b;
            a.x = sX[lo][ks];        a.y = sX[lo][ks + 1];
            b.x = sB[wave][lo][ks];  b.y = sB[wave][lo][ks + 1];
            c = __builtin_amdgcn_wmma_f32_16x16x4_f32(
                    false, a, false, b, (short)0, c, false, false);
        }

        // ---- VALU pipe: L1 accumulation ----
        // packed subtract (0.5 instr/elem, compiler) + add-with-|abs| (1 instr/elem)
        // lane owns (row = i + 8*hi, col = lo) for i = 0..7 (matches C layout)
        #pragma unroll
        for (int kk = 0; kk < KC; kk += 4) {
            const v4f bv = *(const v4f*)&sB[wave][lo][kk];
            #pragma unroll
            for (int i = 0; i < 8; i++) {
                const v4f xv = *(const v4f*)&sX[i + hi * 8][kk];
                v4f d = xv - bv;
                asm("v_add_f32_e64 %0, %0, |%1|" : "+v"(l1a[i]) : "v"(d.x));
                asm("v_add_f32_e64 %0, %0, |%1|" : "+v"(l1b[i]) : "v"(d.y));
                asm("v_add_f32_e64 %0, %0, |%1|" : "+v"(l1c[i]) : "v"(d.z));
                asm("v_add_f32_e64 %0, %0, |%1|" : "+v"(l1d[i]) : "v"(d.w));
            }
        }
    }

    // ---- combine: xd = L1/T0 + sqrt(sq)/T1 ----
    #pragma unroll
    for (int i = 0; i < 8; i++) {
        const int row = i + hi * 8;
        const float l1 = (l1a[i] + l1b[i]) + (l1c[i] + l1d[i]);
        float sq = sNX[row] + sNB[wave * WT + lo] - 2.0f * c[i];
        sq = sq > 0.f ? sq : 0.f;
        const float v = l1 + 0.5f * __builtin_sqrtf(sq);
        xd[(size_t)(b0 + row) * Od + (ow + lo)] = v;
    }
}

// In-place per-row correction: out = -(xd*(1+ALPHA) - rowsum(xd)*ALPHA)
__global__ __launch_bounds__(256) void CDR_rowfix_kernel(float* __restrict__ xd)
{
    const int row = blockIdx.x;
    const int t   = threadIdx.x;                 // 256 threads == Od
    const float v = xd[(size_t)row * Od + t];

    __shared__ float red[Od];
    red[t] = v;
    __syncthreads();
    #pragma unroll
    for (int s = Od / 2; s > 0; s >>= 1) {
        if (t < s) red[t] += red[t + s];
        __syncthreads();
    }
    const float rowsum = red[0];

    xd[(size_t)row * Od + t] = -(v * (1.0f + ALPHA) - rowsum * ALPHA);
}

extern "C" void kernel_launch(void* const* d_in, const int* in_sizes, int n_in,
                              void* d_out, int out_size, void* d_ws, size_t ws_size,
                              hipStream_t stream) {
    (void)n_in; (void)d_ws; (void)ws_size; (void)out_size;
    const float* x     = (const float*)d_in[0];   // [B, D]
    const float* basis = (const float*)d_in[1];   // [O, D]
    float* out = (float*)d_out;                   // [B, O]

    const int B = in_sizes[0] / Dd;               // 1024 for the reference

    dim3 grid1(Od / CT, B / BT);                  // (2, 64)
    CDR_dist_wmma_kernel<<<grid1, 256, 0, stream>>>(x, basis, out);

    CDR_rowfix_kernel<<<B, Od, 0, stream>>>(out); // 1024 blocks, in-place
}